// UntiedConv_59863254171649
// MI455X (gfx1250) — compile-verified
//
#include <hip/hip_runtime.h>

// Untied (locally-connected) conv for MI455X / gfx1250.
// Per output location l: GEMM [M=32 batch] x [K=576] x [N=64 oc], fp32.
// Memory-bound: 151 MB weights (single-use, NT + prefetch) + 8 MB input
// (9x reuse, keep L2-resident) + 8 MB output (single-use, NT) -> ~167 MB
// ~ 7 us floor at 23.3 TB/s. fp32 WMMA (V_WMMA_F32_16X16X4_F32) keeps full
// precision; its issue rate is far below the memory floor.
//
// KCHUNK=144 -> 27.6 KB LDS/block so several blocks fit per CU: more waves
// per SIMD = more outstanding weight loads = closer to HBM saturation.

typedef __attribute__((ext_vector_type(2))) float v2f;
typedef __attribute__((ext_vector_type(8))) float v8f;

#define B_    32
#define C_    64
#define H_    32
#define W_    32
#define OC_   64
#define L_    1024
#define LEN_  576          // C*KH*KW
#define KCHUNK 144         // K staged in LDS per chunk (4 chunks); 144 = 16 ch * 9
// Pair stride in floats. 96 == 32 mod 64 banks -> the two half-waves
// (pairs p and p+1) of a ds_load_b64 A-fragment read disjoint bank halves.
#define PAIR_STRIDE 96
#define PF_ROWS 64         // weight prefetch distance in K-rows (~16 K-steps)

__global__ __launch_bounds__(128)
void untied_conv_wmma_f32(const float* __restrict__ x,
                          const float* __restrict__ w,
                          const float* __restrict__ bias,
                          float* __restrict__ out) {
    __shared__ float A_lds[(KCHUNK / 2) * PAIR_STRIDE];   // 27,648 bytes

    const int l  = blockIdx.x;        // output location
    const int oh = l >> 5;
    const int ow = l & 31;

    const int t    = threadIdx.x;
    const int lane = t & 31;
    const int wave = t >> 5;          // 0..3 -> N-tile
    const int half = lane >> 4;       // 0/1: which K sub-pair of the fragment
    const int col  = lane & 15;       // N within tile / M within tile
    const int n0   = wave * 16;

    const float* wl = w + (size_t)l * LEN_ * OC_;

    v8f acc0 = {};   // batches  0..15
    v8f acc1 = {};   // batches 16..31

    // Fill-phase mapping: 4 threads per batch row m.
    const int fm = t >> 2;            // 0..31
    const int fq = t & 3;
    const float* xm = x + (size_t)fm * C_ * H_ * W_;

    for (int kbase = 0; kbase < LEN_; kbase += KCHUNK) {
        const int cbase = kbase / 9;  // 0,16,32,48 (KCHUNK is a multiple of 9)
        __syncthreads();              // previous chunk fully consumed

        // Gather unf[m][kbase + lk] -> A_lds[(lk>>1)*96 + m*2 + (lk&1)],
        // lk = c_local*9 + kh*3 + kw. Division-free: iterate (kh,kw,c_local).
        #pragma unroll
        for (int kh = 0; kh < 3; ++kh) {
            const int  ih  = oh + kh - 1;
            const bool okh = (unsigned)ih < (unsigned)H_;
            #pragma unroll
            for (int kw = 0; kw < 3; ++kw) {
                const int  iw = ow + kw - 1;
                const bool ok = okh && ((unsigned)iw < (unsigned)W_);
                const int  r  = kh * 3 + kw;
                for (int cl = fq; cl < KCHUNK / 9; cl += 4) {
                    const int lk = cl * 9 + r;
                    float v = 0.0f;
                    if (ok)
                        v = xm[((size_t)(cbase + cl) * H_ + ih) * W_ + iw];
                    A_lds[(lk >> 1) * PAIR_STRIDE + fm * 2 + (lk & 1)] = v;
                }
            }
        }
        __syncthreads();

        // 36 K-steps of 4 over this chunk; 2 wmma per step (both M-tiles).
        #pragma unroll 4
        for (int kk = 0; kk < KCHUNK; kk += 4) {
            // Once per 4-step group: prefetch this wave's weight segments
            // PF_ROWS ahead. Lanes spread over 16 rows x 2 offsets so one
            // instruction covers the group's future 64B segments.
            if ((kk & 15) == 0) {
                const size_t pfrow = (size_t)(kbase + kk + PF_ROWS + (lane & 15));
                __builtin_prefetch(&wl[pfrow * OC_ + n0 + half * 8], 0, 1);
            }
            const int p = (kk >> 1) + half;   // local k-pair for this lane
            // A fragments: v0 <-> k = kk+2*half, v1 <-> k = kk+1+2*half (ISA layout)
            const v2f a0 = *(const v2f*)&A_lds[p * PAIR_STRIDE + col * 2];
            const v2f a1 = *(const v2f*)&A_lds[p * PAIR_STRIDE + (16 + col) * 2];
            // B fragment: rows k = kk+2*half, kk+1+2*half at column n0+col.
            // Weights are single-use -> non-temporal loads keep input hot in L2.
            const int row = kbase + kk + 2 * half;
            v2f bf;
            bf.x = __builtin_nontemporal_load(&wl[(size_t)row * OC_ + n0 + col]);
            bf.y = __builtin_nontemporal_load(&wl[(size_t)(row + 1) * OC_ + n0 + col]);
            acc0 = __builtin_amdgcn_wmma_f32_16x16x4_f32(
                false, a0, false, bf, (short)0, acc0, false, false);
            acc1 = __builtin_amdgcn_wmma_f32_16x16x4_f32(
                false, a1, false, bf, (short)0, acc1, false, false);
        }
    }

    // Epilogue: bias add + store. D layout: vgpr v -> batch m = v + 8*half,
    // lane&15 -> oc = n0+col. Output [B, OC, L] is strided in l -> scatter
    // (unavoidable for a single l; only 8 MB total). Single-use -> NT stores.
    const int   oc = n0 + col;
    const float bv = bias[(size_t)oc * L_ + l];
    float* outp = out + (size_t)oc * L_ + l;   // b = 0 term added below
    #pragma unroll
    for (int v = 0; v < 8; ++v) {
        const int b0 = 8 * half + v;
        const int b1 = 16 + b0;
        __builtin_nontemporal_store(acc0[v] + bv, outp + (size_t)b0 * OC_ * L_);
        __builtin_nontemporal_store(acc1[v] + bv, outp + (size_t)b1 * OC_ * L_);
    }
}

extern "C" void kernel_launch(void* const* d_in, const int* in_sizes, int n_in,
                              void* d_out, int out_size, void* d_ws, size_t ws_size,
                              hipStream_t stream) {
    (void)in_sizes; (void)n_in; (void)d_ws; (void)ws_size; (void)out_size;
    const float* x    = (const float*)d_in[0];
    const float* w    = (const float*)d_in[1];
    const float* bias = (const float*)d_in[2];
    float* out        = (float*)d_out;

    dim3 grid(L_);      // one block per output location
    dim3 block(128);    // 4 waves: one per 16-wide N tile
    untied_conv_wmma_f32<<<grid, block, 0, stream>>>(x, w, bias, out);
}